// SparseFeatureLinear_7189775253943
// MI455X (gfx1250) — compile-verified
//
#include <hip/hip_runtime.h>

// SparseFeatureLinear on MI455X (gfx1250, wave32).
// out[B,16] = x_cont @ W_cont + 1024*bias  +  mask(x_cat) @ (W_cat + bias_row)
// Memory-bound: ~360 MB streamed -> ~15.5 us floor @ 23.3 TB/s. fp32 WMMA 16x16x4.
// MLP strategy: scalar (readfirstlane) wave id + compile-time-constant trip counts
// -> fully unrolled, software-pipelined bodies with ~10 NT loads in flight/wave;
// 512 blocks x 10 waves = 5120 waves ~= 13 MB chip-wide in flight.
// All inner-body addresses are immediate offsets from two base pointers that
// advance once per outer iteration (no per-step 64-bit address VALU).

typedef float v2f __attribute__((ext_vector_type(2)));
typedef float v8f __attribute__((ext_vector_type(8)));

#define BATCH   8192
#define DCONT   1024
#define VCAT    10000
#define OUTD    16

#define NWAVES      10                  // waves per block
#define CAT_STEP    (4 * NWAVES)        // 40 floats per wave per step
#define CAT_OUTER   25
#define CAT_INNER   10                  // 25*10*40 = 10000 exact
#define CONT_WAVES  8
#define CONT_STEP   (4 * CONT_WAVES)    // 32 floats per wave per step
#define CONT_OUTER  8
#define CONT_INNER  4                   // 8*4*32 = 1024 exact

__global__ __launch_bounds__(32 * NWAVES)
void sparse_feature_linear_wmma(const float* __restrict__ x_cont,
                                const float* __restrict__ x_cat,
                                const float* __restrict__ W_cont,
                                const float* __restrict__ W_cat,
                                const float* __restrict__ bias,
                                float* __restrict__ out)
{
    __shared__ float red[NWAVES][8][32];   // partial C tiles, 10 KB

    // readfirstlane: uniform by construction, forces scalar loop control
    const int wave = __builtin_amdgcn_readfirstlane((int)(threadIdx.x >> 5));
    const int lane = threadIdx.x & 31;
    const int n    = lane & 15;   // N column (B/C/D) and M row (A operand)
    const int half = lane >> 4;   // 0: lanes 0-15, 1: lanes 16-31
    const int m0   = blockIdx.x * 16;   // batch tile base row

    const float bn = bias[n];

    // C init: wave 0 carries the DCONT*bias term (same value in every C VGPR,
    // since each C VGPR holds column N = lane%16 of some row M).
    v8f c = {};
    {
        const float cinit = (wave == 0) ? (float)DCONT * bn : 0.0f;
#pragma unroll
        for (int r = 0; r < 8; ++r) c[r] = cinit;
    }

    // ---------------- continuous branch: A = x_cont tile, B = W_cont ----------------
    // waves 0..7 only: 8 waves x 8 outer x 4 unrolled x K4 = 1024
    if (wave < CONT_WAVES) {
        // base pointers; inner offsets are compile-time immediates
        const float* ap = x_cont + (size_t)(m0 + n) * DCONT + wave * 4 + half * 2;
        const float* wp = W_cont + (size_t)(wave * 4 + half * 2) * OUTD + n;
        for (int i = 0; i < CONT_OUTER; ++i) {
#pragma unroll
            for (int u = 0; u < CONT_INNER; ++u) {
                // A 16x4 f32 layout: lane m holds K = k + 2*half + {0,1} in 2 VGPRs.
                v2f a = __builtin_nontemporal_load((const v2f*)(ap + u * CONT_STEP));
                v2f b = { wp[u * CONT_STEP * OUTD],
                          wp[u * CONT_STEP * OUTD + OUTD] };
                c = __builtin_amdgcn_wmma_f32_16x16x4_f32(false, a, false, b,
                                                          (short)0, c, false, false);
            }
            ap += CONT_INNER * CONT_STEP;
            wp += CONT_INNER * CONT_STEP * OUTD;
        }
    }

    // ---------------- categorical branch: A = (x_cat != 0), B = W_cat + bias --------
    // all 10 waves: 10 waves x 25 outer x 10 unrolled x K4 = 10000
    {
        const float* ap = x_cat + (size_t)(m0 + n) * VCAT + wave * 4 + half * 2;
        const float* wp = W_cat + (size_t)(wave * 4 + half * 2) * OUTD + n;
        for (int i = 0; i < CAT_OUTER; ++i) {
            // keep the HBM stream ahead of the serial WMMA accumulation chain
            __builtin_prefetch(ap + 2 * CAT_INNER * CAT_STEP, 0, 1);  // 2 bodies ahead
#pragma unroll
            for (int u = 0; u < CAT_INNER; ++u) {
                v2f a = __builtin_nontemporal_load((const v2f*)(ap + u * CAT_STEP));
                a.x = (a.x != 0.0f) ? 1.0f : 0.0f;          // multi-hot mask on the fly
                a.y = (a.y != 0.0f) ? 1.0f : 0.0f;
                v2f b = { wp[u * CAT_STEP * OUTD] + bn,     // fold count*bias into B
                          wp[u * CAT_STEP * OUTD + OUTD] + bn };
                c = __builtin_amdgcn_wmma_f32_16x16x4_f32(false, a, false, b,
                                                          (short)0, c, false, false);
            }
            ap += CAT_INNER * CAT_STEP;
            wp += CAT_INNER * CAT_STEP * OUTD;
        }
    }

    // ---------------- cross-wave K reduction through LDS ----------------------------
#pragma unroll
    for (int r = 0; r < 8; ++r) red[wave][r][lane] = c[r];
    __syncthreads();

    if (wave == 0) {
#pragma unroll
        for (int r = 0; r < 8; ++r) {
            float s = 0.0f;
#pragma unroll
            for (int w = 0; w < NWAVES; ++w) s += red[w][r][lane];
            // C/D layout: lanes 0-15 -> row r, lanes 16-31 -> row r+8, col n.
            out[(size_t)(m0 + r + half * 8) * OUTD + n] = s;
        }
    }
}

extern "C" void kernel_launch(void* const* d_in, const int* in_sizes, int n_in,
                              void* d_out, int out_size, void* d_ws, size_t ws_size,
                              hipStream_t stream) {
    (void)in_sizes; (void)n_in; (void)d_ws; (void)ws_size; (void)out_size;
    const float* x_cont = (const float*)d_in[0];
    const float* x_cat  = (const float*)d_in[1];
    const float* W_cont = (const float*)d_in[2];
    const float* W_cat  = (const float*)d_in[3];
    const float* bias   = (const float*)d_in[4];
    float*       out    = (float*)d_out;

    dim3 grid(BATCH / 16);        // 512 tiles
    dim3 block(32 * NWAVES);      // 10 wave32s per block
    sparse_feature_linear_wmma<<<grid, block, 0, stream>>>(
        x_cont, x_cat, W_cont, W_cat, bias, out);
}